// ProbAttention_73083163508986
// MI455X (gfx1250) — compile-verified
//
#include <hip/hip_runtime.h>
#include <math.h>

typedef __attribute__((ext_vector_type(2))) float v2f;
typedef __attribute__((ext_vector_type(8))) float v8f;

#define B_  8
#define L_  2048
#define H_  8
#define D_  64
#define SK_ 40
#define NT_ 40

// ---------------------------------------------------------------------------
// Kernel 1: sparsity measure M + top-40 selection, one workgroup per (b,h).
// ---------------------------------------------------------------------------
__global__ __launch_bounds__(256) void prob_mtop_kernel(
    const float* __restrict__ q, const float* __restrict__ k,
    const int* __restrict__ isamp, int* __restrict__ mtop)
{
    const int bh = blockIdx.x;
    const int b  = bh >> 3;
    const int h  = bh & 7;
    const int t  = threadIdx.x;

    __shared__ float Msh[L_];
    __shared__ float rv[256];
    __shared__ int   ri[256];
    __shared__ int   sel[NT_];

    // Phase 1: M[l] = max_s(q.k_s) - sum_s(q.k_s)/L
    for (int l = t; l < L_; l += 256) {
        const float4* qrow = (const float4*)(q + ((size_t)(b * L_ + l) * H_ + h) * D_);
        float4 qr[16];
#pragma unroll
        for (int i = 0; i < 16; ++i) qr[i] = qrow[i];
        float mx = -3.0e38f, sm = 0.0f;
        for (int s = 0; s < SK_; ++s) {
            const int idx = isamp[l * SK_ + s];
            const float4* krow = (const float4*)(k + ((size_t)(b * L_ + idx) * H_ + h) * D_);
            float dot = 0.0f;
#pragma unroll
            for (int i = 0; i < 16; ++i) {
                float4 kv = krow[i];
                dot += qr[i].x * kv.x + qr[i].y * kv.y + qr[i].z * kv.z + qr[i].w * kv.w;
            }
            mx = fmaxf(mx, dot);
            sm += dot;
        }
        Msh[l] = mx - sm * (1.0f / (float)L_);
    }
    __syncthreads();

    // Phase 2: iterative top-40 (ties -> lowest index, like jax top_k)
    for (int r = 0; r < NT_; ++r) {
        float bv = -3.0e38f;
        int   bi = 0x7fffffff;
        for (int l = t; l < L_; l += 256) {
            float v = Msh[l];
            if (v > bv || (v == bv && l < bi)) { bv = v; bi = l; }
        }
        rv[t] = bv; ri[t] = bi;
        __syncthreads();
        for (int off = 128; off > 0; off >>= 1) {
            if (t < off) {
                float ov = rv[t + off]; int oi = ri[t + off];
                if (ov > rv[t] || (ov == rv[t] && oi < ri[t])) { rv[t] = ov; ri[t] = oi; }
            }
            __syncthreads();
        }
        if (t == 0) { sel[r] = ri[0]; Msh[ri[0]] = -3.0e38f; }
        __syncthreads();
    }
    if (t < NT_) mtop[bh * NT_ + t] = sel[t];
}

// ---------------------------------------------------------------------------
// Kernel 2: flash-style attention for the 40 selected queries + mean(V)
// broadcast, one workgroup (4 waves) per (b,h). Waves 0..2 own 16-row
// M-tiles (rows 40..47 are zero padding). Uses V_WMMA_F32_16X16X4_F32.
// ---------------------------------------------------------------------------
__global__ __launch_bounds__(128) void prob_attn_kernel(
    const float* __restrict__ q, const float* __restrict__ k,
    const float* __restrict__ v, const int* __restrict__ mtop,
    float* __restrict__ out)
{
    const int bh = blockIdx.x;
    const int b  = bh >> 3;
    const int h  = bh & 7;
    const int t  = threadIdx.x;   // 0..127

    __shared__ __align__(16) float qred[48 * D_];
    __shared__ __align__(16) float upd[48 * D_];
    __shared__ __align__(16) float meanv[D_];
    __shared__ float vpart[2 * D_];
    __shared__ int   flags[L_];
    __shared__ int   mt[NT_];
    __shared__ __align__(16) float pt[3][16 * 16];

    if (t < NT_) mt[t] = mtop[bh * NT_ + t];
    for (int l = t; l < L_; l += 128) flags[l] = 0;
    __syncthreads();
    if (t < NT_) flags[mt[t]] = t + 1;

    // q_reduce -> LDS (48 rows, padded with zeros)
    for (int idx = t; idx < 48 * D_; idx += 128) {
        const int row = idx >> 6, col = idx & 63;
        qred[idx] = (row < NT_)
            ? q[((size_t)(b * L_ + mt[row]) * H_ + h) * D_ + col] : 0.0f;
    }

    // column sums of V for mean context
    {
        const int col = t & 63, half = t >> 6;
        float s = 0.0f;
        for (int l = half; l < L_; l += 2)
            s += v[((size_t)(b * L_ + l) * H_ + h) * D_ + col];
        vpart[half * D_ + col] = s;
    }
    __syncthreads();
    if (t < D_) meanv[t] = (vpart[t] + vpart[D_ + t]) * (1.0f / (float)L_);
    __syncthreads();

    const int wave = t >> 5;
    const int ln   = t & 31;

    if (wave < 3) {
        const int h4 = ln >> 4;     // half-wave: 0 or 1
        const int nl = ln & 15;
        const int qrow = wave * 16 + nl;

        // A frags for GEMM1 (32-bit 16x4 A layout: lanes<16 hold K=2h4+{0,1})
        v2f qa[16];
#pragma unroll
        for (int j = 0; j < 16; ++j)
            qa[j] = *(const v2f*)&qred[qrow * D_ + j * 4 + 2 * h4];

        v8f acc[4] = {};
        float mrun[8], lrun[8];
#pragma unroll
        for (int r = 0; r < 8; ++r) { mrun[r] = -3.0e38f; lrun[r] = 0.0f; }

        const float* kbase = k + ((size_t)(b * L_) * H_ + h) * D_;
        const float* vbase = v + ((size_t)(b * L_) * H_ + h) * D_;
        float* ptw = &pt[wave][0];
        const float LOG2E = 1.4426950408889634f;

        for (int n0 = 0; n0 < L_; n0 += 16) {
            // ---- GEMM1: scores tile (16x16) = q_tile (16x64) . K_chunk^T ----
            v8f c = {};
            const float* krow = kbase + (size_t)(n0 + nl) * (H_ * D_) + 2 * h4;
#pragma unroll
            for (int j = 0; j < 16; ++j) {
                v2f bfr = *(const v2f*)(krow + j * 4);
                c = __builtin_amdgcn_wmma_f32_16x16x4_f32(
                        false, qa[j], false, bfr, (short)0, c, false, false);
            }

            // ---- online softmax (rows = 16-lane groups in C layout) ----
            float sreg[8], preg[8];
#pragma unroll
            for (int r = 0; r < 8; ++r) sreg[r] = c[r] * 0.125f;  // 1/sqrt(64)
#pragma unroll
            for (int r = 0; r < 8; ++r) {
                float m = sreg[r];
                m = fmaxf(m, __shfl_xor(m, 1, 32));
                m = fmaxf(m, __shfl_xor(m, 2, 32));
                m = fmaxf(m, __shfl_xor(m, 4, 32));
                m = fmaxf(m, __shfl_xor(m, 8, 32));
                const float mn = fmaxf(mrun[r], m);
                const float p  = exp2f((sreg[r] - mn) * LOG2E);
                float ps = p;
                ps += __shfl_xor(ps, 1, 32);
                ps += __shfl_xor(ps, 2, 32);
                ps += __shfl_xor(ps, 4, 32);
                ps += __shfl_xor(ps, 8, 32);
                const float scale = exp2f((mrun[r] - mn) * LOG2E);
                lrun[r] = lrun[r] * scale + ps;
                mrun[r] = mn;
                preg[r] = p;
                acc[0][r] *= scale; acc[1][r] *= scale;
                acc[2][r] *= scale; acc[3][r] *= scale;
            }

            // ---- transpose P (C layout -> A layout) through per-wave LDS ----
#pragma unroll
            for (int r = 0; r < 8; ++r) ptw[(r + 8 * h4) * 16 + nl] = preg[r];
            v2f a2[4];
#pragma unroll
            for (int j = 0; j < 4; ++j)
                a2[j] = *(const v2f*)&ptw[nl * 16 + j * 4 + 2 * h4];

            // ---- GEMM2: acc (16x64) += P (16x16) . V_chunk (16x64) ----
#pragma unroll
            for (int nt = 0; nt < 4; ++nt) {
                const int d = nt * 16 + nl;
                const float* vb = vbase + ((size_t)n0 + 2 * h4) * (H_ * D_) + d;
#pragma unroll
                for (int j = 0; j < 4; ++j) {
                    v2f bfr;
                    bfr.x = vb[(size_t)(j * 4 + 0) * (H_ * D_)];
                    bfr.y = vb[(size_t)(j * 4 + 1) * (H_ * D_)];
                    acc[nt] = __builtin_amdgcn_wmma_f32_16x16x4_f32(
                                  false, a2[j], false, bfr, (short)0, acc[nt],
                                  false, false);
                }
            }
        }

        // normalize and park the update rows in LDS
#pragma unroll
        for (int r = 0; r < 8; ++r) {
            const float inv = 1.0f / lrun[r];
            const int row = wave * 16 + r + 8 * h4;
#pragma unroll
            for (int nt = 0; nt < 4; ++nt)
                upd[row * D_ + nt * 16 + nl] = acc[nt][r] * inv;
        }
    }
    __syncthreads();

    // ---- write output: mean(V) everywhere, update at selected rows ----
    for (int idx = t; idx < L_ * (D_ / 4); idx += 128) {
        const int row = idx >> 4;
        const int col = (idx & 15) * 4;
        const int u   = flags[row];
        float4 val = u ? *(const float4*)&upd[(u - 1) * D_ + col]
                       : *(const float4*)&meanv[col];
        *(float4*)&out[(((size_t)bh) * L_ + row) * D_ + col] = val;
    }
}

// ---------------------------------------------------------------------------
extern "C" void kernel_launch(void* const* d_in, const int* in_sizes, int n_in,
                              void* d_out, int out_size, void* d_ws, size_t ws_size,
                              hipStream_t stream) {
    (void)in_sizes; (void)n_in; (void)out_size; (void)ws_size;
    const float* q     = (const float*)d_in[0];
    const float* k     = (const float*)d_in[1];
    const float* v     = (const float*)d_in[2];
    const int*   isamp = (const int*)d_in[3];
    float*       out   = (float*)d_out;
    int*         mtop  = (int*)d_ws;   // B*H*40 ints = 10 KB

    prob_mtop_kernel<<<dim3(B_ * H_), dim3(256), 0, stream>>>(q, k, isamp, mtop);
    prob_attn_kernel<<<dim3(B_ * H_), dim3(128), 0, stream>>>(q, k, v, mtop, out);
}